// GCNEncoder_38749194944633
// MI455X (gfx1250) — compile-verified
//
#include <hip/hip_runtime.h>

// Problem constants (match reference; also derived from in_sizes at launch)
#define FEAT 128
#define EPS_BN 1e-5f

typedef float v2f __attribute__((ext_vector_type(2)));
typedef float v8f __attribute__((ext_vector_type(8)));

// ---------------------------------------------------------------------------
// f32 WMMA 16x16x4 wrapper:  D = A(16x4) * B(4x16) + C(16x16), fp32 throughout
// ---------------------------------------------------------------------------
__device__ __forceinline__ v8f wmma_f32_k4(v2f a, v2f b, v8f c) {
  return __builtin_amdgcn_wmma_f32_16x16x4_f32(
      /*neg_a=*/false, a, /*neg_b=*/false, b,
      /*c_mod=*/(short)0, c, /*reuse_a=*/false, /*reuse_b=*/false);
}

// ---------------------------------------------------------------------------
// GEMM: C[n x 128] = A[n x 128] @ W[128 x 128]   (row-major, no bias)
// One block = 16 output rows; wave w handles output cols [16w, 16w+16).
// ---------------------------------------------------------------------------
__global__ __launch_bounds__(256) void gemm128_wmma_kernel(
    const float* __restrict__ A, const float* __restrict__ W,
    float* __restrict__ C, int n) {
  const int lane = threadIdx.x & 31;
  const int wave = threadIdx.x >> 5;       // 0..7  -> N tile
  const int half = lane >> 4;              // 0/1
  const int r    = lane & 15;
  const int m0   = blockIdx.x * 16;
  const int col  = wave * 16 + r;
  const int rowL = min(m0 + r, n - 1);     // clamp for (unused) tail rows

  v8f c = {};
  for (int k = 0; k < FEAT; k += 4) {
    const int kb = k + 2 * half;           // A lanes0-15: K0/K1, lanes16-31: K2/K3
    v2f a = *(const v2f*)(A + (long)rowL * FEAT + kb);
    v2f b;
    b.x = W[(long)kb * FEAT + col];
    b.y = W[(long)(kb + 1) * FEAT + col];
    c = wmma_f32_k4(a, b, c);
  }
  // C layout: lane l, vgpr v -> row m0 + 8*(l>>4) + v, col = wave*16 + (l&15)
  const int rbase = m0 + 8 * half;
#pragma unroll
  for (int v = 0; v < 8; ++v)
    if (rbase + v < n) C[(long)(rbase + v) * FEAT + col] = c[v];
}

// ---------------------------------------------------------------------------
// Final GEMM: out[n x 128] = [H | relu(dist*ad+cd) | relu(deg*ag+cg)] @ Wm + bm
// Side features are generated in-register (never materialized).
// ---------------------------------------------------------------------------
__global__ __launch_bounds__(256) void final_gemm_wmma_kernel(
    const float* __restrict__ H, const float* __restrict__ dist,
    const float* __restrict__ degf,
    const float* __restrict__ ad, const float* __restrict__ cd,
    const float* __restrict__ ag, const float* __restrict__ cg,
    const float* __restrict__ Wm, const float* __restrict__ bm,
    float* __restrict__ out, int n) {
  const int lane = threadIdx.x & 31;
  const int wave = threadIdx.x >> 5;
  const int half = lane >> 4;
  const int r    = lane & 15;
  const int m0   = blockIdx.x * 16;
  const int col  = wave * 16 + r;
  const int rowL = min(m0 + r, n - 1);
  const float dv = dist[rowL];
  const float gv = degf[rowL];

  v8f c = {};
  for (int k = 0; k < 3 * FEAT; k += 4) {
    const int kb = k + 2 * half;           // kb,kb+1 stay inside one 4-aligned group
    v2f a;
    if (kb < FEAT) {
      a = *(const v2f*)(H + (long)rowL * FEAT + kb);
    } else if (kb < 2 * FEAT) {
      const int j = kb - FEAT;
      a.x = fmaxf(dv * ad[j]     + cd[j],     0.0f);
      a.y = fmaxf(dv * ad[j + 1] + cd[j + 1], 0.0f);
    } else {
      const int j = kb - 2 * FEAT;
      a.x = fmaxf(gv * ag[j]     + cg[j],     0.0f);
      a.y = fmaxf(gv * ag[j + 1] + cg[j + 1], 0.0f);
    }
    v2f b;
    b.x = Wm[(long)kb * FEAT + col];
    b.y = Wm[(long)(kb + 1) * FEAT + col];
    c = wmma_f32_k4(a, b, c);
  }
  const float bias = bm[col];
  const int rbase = m0 + 8 * half;
#pragma unroll
  for (int v = 0; v < 8; ++v)
    if (rbase + v < n) out[(long)(rbase + v) * FEAT + col] = c[v] + bias;
}

// ---------------------------------------------------------------------------
// Graph normalization helpers
// ---------------------------------------------------------------------------
__global__ void init_deg_kernel(float* deg, int n) {
  int i = blockIdx.x * blockDim.x + threadIdx.x;
  if (i < n) deg[i] = 1.0f;                // self-loop weight 1
}

__global__ void deg_accum_kernel(const int* __restrict__ dst,
                                 const float* __restrict__ ew,
                                 float* __restrict__ deg, int e) {
  int i = blockIdx.x * blockDim.x + threadIdx.x;
  if (i < e) unsafeAtomicAdd(&deg[dst[i]], ew[i]);
}

__global__ void dinv_kernel(const float* __restrict__ deg,
                            float* __restrict__ dinv, int n) {
  int i = blockIdx.x * blockDim.x + threadIdx.x;
  if (i < n) {
    float d = deg[i];
    dinv[i] = (d > 0.0f) ? rsqrtf(d) : 0.0f;
  }
}

// out[i][j] = bias[j] + dinv[i]^2 * h[i][j]    (self-loop term + conv bias)
__global__ void selfloop_init_kernel(const float* __restrict__ h,
                                     const float* __restrict__ dinv,
                                     const float* __restrict__ bias,
                                     float* __restrict__ out, int n) {
  long i = (long)blockIdx.x * blockDim.x + threadIdx.x;
  long total = (long)n * FEAT;
  if (i < total) {
    int col = (int)(i & (FEAT - 1));
    long row = i >> 7;
    float di = dinv[row];
    out[i] = bias[col] + di * di * h[i];
  }
}

// One wave per edge; lane handles 4 features (float4 gather + 4 f32 atomics)
__global__ __launch_bounds__(256) void scatter_kernel(
    const float* __restrict__ h, const int* __restrict__ src,
    const int* __restrict__ dst, const float* __restrict__ ew,
    const float* __restrict__ dinv, float* __restrict__ out, int e) {
  int w = (int)(((long)blockIdx.x * blockDim.x + threadIdx.x) >> 5);
  int lane = threadIdx.x & 31;
  if (w >= e) return;
  int s = src[w], d = dst[w];
  float coef = dinv[s] * ew[w] * dinv[d];
  float4 v = *(const float4*)(h + (long)s * FEAT + lane * 4);
  float* op = out + (long)d * FEAT + lane * 4;
  unsafeAtomicAdd(op + 0, v.x * coef);
  unsafeAtomicAdd(op + 1, v.y * coef);
  unsafeAtomicAdd(op + 2, v.z * coef);
  unsafeAtomicAdd(op + 3, v.w * coef);
}

// ---------------------------------------------------------------------------
// BatchNorm statistics and application
// ---------------------------------------------------------------------------
__global__ void zero_stats_kernel(float* p, int n) {
  int i = blockIdx.x * blockDim.x + threadIdx.x;
  if (i < n) p[i] = 0.0f;
}

__global__ __launch_bounds__(256) void colstats_kernel(
    const float* __restrict__ h, float* __restrict__ sum,
    float* __restrict__ sumsq, int n) {
  __shared__ float red[2][256];
  const int col  = threadIdx.x & (FEAT - 1);
  const int part = threadIdx.x >> 7;       // 0/1
  float s = 0.0f, q = 0.0f;
  const int base = blockIdx.x * 256;
  for (int i = part; i < 256; i += 2) {
    int row = base + i;
    if (row < n) {
      float v = h[(long)row * FEAT + col];
      s += v;
      q += v * v;
    }
  }
  red[0][threadIdx.x] = s;
  red[1][threadIdx.x] = q;
  __syncthreads();
  if (threadIdx.x < FEAT) {
    unsafeAtomicAdd(&sum[col],   red[0][threadIdx.x] + red[0][threadIdx.x + 128]);
    unsafeAtomicAdd(&sumsq[col], red[1][threadIdx.x] + red[1][threadIdx.x + 128]);
  }
}

__global__ void bn_coeffs_kernel(const float* __restrict__ sum,
                                 const float* __restrict__ sumsq,
                                 const float* __restrict__ g,
                                 const float* __restrict__ b,
                                 float* __restrict__ a, float* __restrict__ c,
                                 float invn) {
  int j = threadIdx.x;
  float mu  = sum[j] * invn;
  float var = sumsq[j] * invn - mu * mu;
  float aj  = g[j] * rsqrtf(var + EPS_BN);
  a[j] = aj;
  c[j] = b[j] - mu * aj;
}

__global__ void bn_relu_kernel(float* __restrict__ h, const float* __restrict__ a,
                               const float* __restrict__ c, int n) {
  long i = (long)blockIdx.x * blockDim.x + threadIdx.x;
  if (i < (long)n * FEAT) {
    int col = (int)(i & (FEAT - 1));
    h[i] = fmaxf(h[i] * a[col] + c[col], 0.0f);
  }
}

// Scalar mean/var of the two 1-D side inputs
__global__ __launch_bounds__(256) void scalar_stats_kernel(
    const float* __restrict__ dist, const float* __restrict__ degf,
    float* __restrict__ scal, int n) {
  __shared__ float red[4][256];
  int idx = blockIdx.x * 256 + threadIdx.x;
  float d = 0.0f, g = 0.0f;
  if (idx < n) { d = dist[idx]; g = degf[idx]; }
  red[0][threadIdx.x] = d; red[1][threadIdx.x] = d * d;
  red[2][threadIdx.x] = g; red[3][threadIdx.x] = g * g;
  __syncthreads();
  for (int st = 128; st > 0; st >>= 1) {
    if (threadIdx.x < st)
#pragma unroll
      for (int j = 0; j < 4; ++j) red[j][threadIdx.x] += red[j][threadIdx.x + st];
    __syncthreads();
  }
  if (threadIdx.x == 0)
#pragma unroll
    for (int j = 0; j < 4; ++j) unsafeAtomicAdd(&scal[j], red[j][0]);
}

// Side branch: relu(bn(x @ W + b)) collapses to relu(x*a[j] + c[j])
// (the linear bias cancels against the BN mean subtraction)
__global__ void side_coeffs_kernel(const float* __restrict__ scal,
                                   const float* __restrict__ Wd,
                                   const float* __restrict__ gd,
                                   const float* __restrict__ bdn,
                                   const float* __restrict__ Wg,
                                   const float* __restrict__ gg,
                                   const float* __restrict__ bgn,
                                   float* __restrict__ ad, float* __restrict__ cd,
                                   float* __restrict__ ag, float* __restrict__ cg,
                                   float invn) {
  int j = threadIdx.x;
  float mud  = scal[0] * invn;
  float vard = scal[1] * invn - mud * mud;
  float mug  = scal[2] * invn;
  float varg = scal[3] * invn - mug * mug;
  float w    = Wd[j];
  float adj  = w * rsqrtf(vard * w * w + EPS_BN) * gd[j];
  ad[j] = adj;
  cd[j] = bdn[j] - mud * adj;
  w = Wg[j];
  float agj = w * rsqrtf(varg * w * w + EPS_BN) * gg[j];
  ag[j] = agj;
  cg[j] = bgn[j] - mug * agj;
}

// ---------------------------------------------------------------------------
extern "C" void kernel_launch(void* const* d_in, const int* in_sizes, int n_in,
                              void* d_out, int out_size, void* d_ws, size_t ws_size,
                              hipStream_t stream) {
  const float* x    = (const float*)d_in[0];
  const int*   ei   = (const int*)d_in[1];
  const float* ew   = (const float*)d_in[2];
  const float* dist = (const float*)d_in[3];
  const float* degf = (const float*)d_in[4];
  const float* W1   = (const float*)d_in[5];
  const float* b1   = (const float*)d_in[6];
  const float* W2   = (const float*)d_in[7];
  const float* b2   = (const float*)d_in[8];
  const float* bn1g = (const float*)d_in[9];
  const float* bn1b = (const float*)d_in[10];
  const float* bn2g = (const float*)d_in[11];
  const float* bn2b = (const float*)d_in[12];
  const float* Wd   = (const float*)d_in[13];
  const float* bndg = (const float*)d_in[15];
  const float* bndb = (const float*)d_in[16];
  const float* Wg   = (const float*)d_in[17];
  const float* bngg = (const float*)d_in[19];
  const float* bngb = (const float*)d_in[20];
  const float* Wm   = (const float*)d_in[21];
  const float* bm   = (const float*)d_in[22];

  const int n = in_sizes[0] / FEAT;        // 100000
  const int e = in_sizes[1] / 2;           // 1600000
  const int* src = ei;
  const int* dst = ei + e;

  // Workspace layout (floats). Needs 2*n*128 + 2n + ~1.2K floats (~103 MB).
  float* ws   = (float*)d_ws;
  float* bufA = ws;                         // GEMM outputs (h = x@W)
  float* bufB = bufA + (long)n * FEAT;      // conv outputs / BN'ed features
  float* deg  = bufB + (long)n * FEAT;
  float* dinv = deg + n;
  float* sum  = dinv + n;                   // [128]
  float* sumsq = sum + FEAT;                // [128]
  float* scal  = sumsq + FEAT;              // [4]
  float* aco   = scal + 4;                  // [128] BN scale
  float* cco   = aco + FEAT;                // [128] BN shift
  float* ad    = cco + FEAT;
  float* cd    = ad + FEAT;
  float* ag    = cd + FEAT;
  float* cg    = ag + FEAT;

  const int  TB   = 256;
  const int  nbN  = (n + TB - 1) / TB;                 // 391
  const int  nbE  = (e + TB - 1) / TB;                 // 6250
  const long nelL = (long)n * FEAT;
  const int  nbEl = (int)((nelL + TB - 1) / TB);       // 50000
  const int  nbMt = (n + 15) / 16;                     // 6250 (WMMA M tiles)
  const int  nbSc = (e + 7) / 8;                       // 200000 (8 edge-waves/block)
  const float invn = 1.0f / (float)n;

  // --- degree / normalization ---
  init_deg_kernel<<<nbN, TB, 0, stream>>>(deg, n);
  zero_stats_kernel<<<2, TB, 0, stream>>>(sum, 2 * FEAT + 4);  // sum,sumsq,scal
  deg_accum_kernel<<<nbE, TB, 0, stream>>>(dst, ew, deg, e);
  dinv_kernel<<<nbN, TB, 0, stream>>>(deg, dinv, n);

  // --- layer 1: GCNConv -> BN -> ReLU ---
  gemm128_wmma_kernel<<<nbMt, TB, 0, stream>>>(x, W1, bufA, n);
  selfloop_init_kernel<<<nbEl, TB, 0, stream>>>(bufA, dinv, b1, bufB, n);
  scatter_kernel<<<nbSc, TB, 0, stream>>>(bufA, src, dst, ew, dinv, bufB, e);
  colstats_kernel<<<nbN, TB, 0, stream>>>(bufB, sum, sumsq, n);
  bn_coeffs_kernel<<<1, FEAT, 0, stream>>>(sum, sumsq, bn1g, bn1b, aco, cco, invn);
  bn_relu_kernel<<<nbEl, TB, 0, stream>>>(bufB, aco, cco, n);

  // --- layer 2: GCNConv -> BN -> ReLU ---
  zero_stats_kernel<<<2, TB, 0, stream>>>(sum, 2 * FEAT + 4);
  gemm128_wmma_kernel<<<nbMt, TB, 0, stream>>>(bufB, W2, bufA, n);
  selfloop_init_kernel<<<nbEl, TB, 0, stream>>>(bufA, dinv, b2, bufB, n);
  scatter_kernel<<<nbSc, TB, 0, stream>>>(bufA, src, dst, ew, dinv, bufB, e);
  colstats_kernel<<<nbN, TB, 0, stream>>>(bufB, sum, sumsq, n);
  bn_coeffs_kernel<<<1, FEAT, 0, stream>>>(sum, sumsq, bn2g, bn2b, aco, cco, invn);
  bn_relu_kernel<<<nbEl, TB, 0, stream>>>(bufB, aco, cco, n);

  // --- side branches (collapsed to per-column affine + relu) ---
  scalar_stats_kernel<<<nbN, TB, 0, stream>>>(dist, degf, scal, n);
  side_coeffs_kernel<<<1, FEAT, 0, stream>>>(scal, Wd, bndg, bndb,
                                             Wg, bngg, bngb,
                                             ad, cd, ag, cg, invn);

  // --- fused final GEMM: [h | df | gf] @ Wm + bm ---
  final_gemm_wmma_kernel<<<nbMt, TB, 0, stream>>>(
      bufB, dist, degf, ad, cd, ag, cg, Wm, bm, (float*)d_out, n);
}